// CPI_CLS_49478023250092
// MI455X (gfx1250) — compile-verified
//
#include <hip/hip_runtime.h>
#include <hip/hip_bf16.h>

// ---------------------------------------------------------------------------
// CPI model for MI455X (gfx1250, wave32, WMMA)
//
// Phase 1 (GNN, dominant):   xs += adjacency[4096x4096 f32] @ hs[4096x16]
//   -> V_WMMA_F32_16X16X4_F32, 256 blocks x 8 waves, split-K=512/wave,
//      LDS reduction of partial C tiles (deterministic, no float atomics).
// Phase 2 (CNN): 3x 23x23 conv over [65536,10] image, LDS-staged
//   (async GLOBAL_LOAD_ASYNC_TO_LDS_B128 staging when toolchain provides it).
// Phase 3: attention pooling + fusion MLP (tiny, VALU).
// ---------------------------------------------------------------------------

typedef float v8f __attribute__((ext_vector_type(8)));
typedef float v2f __attribute__((ext_vector_type(2)));
typedef int   v4i __attribute__((ext_vector_type(4)));

#define DPAD   16
#define NATOMS 4096
#define LWORDS 65536

#if defined(__has_builtin)
#if __has_builtin(__builtin_amdgcn_global_load_async_to_lds_b128)
#define HAVE_ASYNC_LDS 1
#endif
#endif

typedef __attribute__((address_space(1))) v4i* gptr_v4i;
typedef __attribute__((address_space(3))) v4i* lptr_v4i;

__device__ __forceinline__ void wait_asynccnt0() {
#if defined(HAVE_ASYNC_LDS)
#if __has_builtin(__builtin_amdgcn_s_wait_asynccnt)
  __builtin_amdgcn_s_wait_asynccnt(0);
#else
  asm volatile("s_wait_asynccnt 0x0" ::: "memory");
#endif
#endif
}

// ---------------------------------------------------------------------------
// Embedding gathers (pad D=10 -> 16 with zeros)
// ---------------------------------------------------------------------------
__global__ void embed_fp_kernel(const int* __restrict__ fp,
                                const float* __restrict__ emb,
                                float* __restrict__ xs) {
  int gid = blockIdx.x * blockDim.x + threadIdx.x;   // NATOMS*16
  int i = gid >> 4, d = gid & 15;
  xs[gid] = (d < 10) ? emb[fp[i] * 10 + d] : 0.0f;
}

__global__ void embed_word_kernel(const int* __restrict__ words,
                                  const float* __restrict__ emb,
                                  float* __restrict__ img) {
  int gid = blockIdx.x * blockDim.x + threadIdx.x;   // LWORDS*16
  int i = gid >> 4, d = gid & 15;
  img[gid] = (d < 10) ? emb[words[i] * 10 + d] : 0.0f;
}

// ---------------------------------------------------------------------------
// hs = relu(xs @ W^T + b), [4096,16] (cols 10..15 zero)
// ---------------------------------------------------------------------------
__global__ void gnn_hs_kernel(const float* __restrict__ xs,
                              const float* __restrict__ W,
                              const float* __restrict__ b,
                              float* __restrict__ hs) {
  int gid = blockIdx.x * blockDim.x + threadIdx.x;   // NATOMS*16
  int i = gid >> 4, d = gid & 15;
  float v = 0.0f;
  if (d < 10) {
    float acc = b[d];
    const float* xr = xs + i * DPAD;
    const float* wr = W + d * 10;
#pragma unroll
    for (int e = 0; e < 10; ++e) acc += xr[e] * wr[e];
    v = fmaxf(acc, 0.0f);
  }
  hs[gid] = v;
}

// ---------------------------------------------------------------------------
// xs += A @ hs  via V_WMMA_F32_16X16X4_F32
// grid = 256 blocks (one 16-row C tile each), 8 waves split K=4096 into 512
// ---------------------------------------------------------------------------
__global__ __launch_bounds__(256) void gnn_gemm_kernel(
    const float* __restrict__ A,     // [4096,4096] row-major
    const float* __restrict__ hs,    // [4096,16]   row-major (B matrix)
    float* __restrict__ xs) {        // [4096,16]   accumulated in place
  __shared__ float shC[8 * 256];     // 8 partial 16x16 tiles

  const int tid  = threadIdx.x;
  const int wave = tid >> 5;
  const int lane = tid & 31;
  const int half = lane >> 4;        // 0: lanes 0-15, 1: lanes 16-31
  const int l16  = lane & 15;
  const int tileBase = blockIdx.x * 16;      // first M row of this tile
  const int kBase    = wave * 512;           // this wave's K range
  const int koff     = 2 * half;             // K offset for upper half-wave

  // A fragment source: lane l16 owns matrix row (tileBase+l16); per K-step it
  // loads 2 consecutive f32 at column kBase + koff + k  (ISA 16x4 f32 layout).
  const float* arow = A + (size_t)(tileBase + l16) * NATOMS + kBase + koff;

  v8f c = {};                                // partial accumulator (C=0)
  for (int k0 = 0; k0 < 512; k0 += 128) {    // one prefetch per 512B chunk
    if (k0 < 384)
      __builtin_prefetch(arow + k0 + 128, 0, 1);   // global_prefetch_b8
    for (int k = k0; k < k0 + 128; k += 4) {
      v2f a = *(const v2f*)(arow + k);
      v2f b;
      // B 4x16 layout: vgpr0 = row (k+koff), vgpr1 = row (k+koff+1)
      b.x = hs[(kBase + k + koff)     * DPAD + l16];
      b.y = hs[(kBase + k + koff + 1) * DPAD + l16];
      c = __builtin_amdgcn_wmma_f32_16x16x4_f32(
          /*neg_a=*/false, a, /*neg_b=*/false, b,
          /*c_mod=*/(short)0, c, /*reuse_a=*/false, /*reuse_b=*/false);
    }
  }

  // Spill partial C tile to LDS in natural (row,col) order.
  // C/D layout: VGPR r -> M=r (lanes 0-15) / M=8+r (lanes 16-31), N=lane%16.
#pragma unroll
  for (int r = 0; r < 8; ++r)
    shC[wave * 256 + (r + 8 * half) * 16 + l16] = c[r];
  __syncthreads();

  // Reduce the 8 split-K partials and accumulate into xs.
  int row = tid >> 4, col = tid & 15;
  float acc = xs[(tileBase + row) * DPAD + col];
#pragma unroll
  for (int w = 0; w < 8; ++w) acc += shC[w * 256 + tid];
  xs[(tileBase + row) * DPAD + col] = acc;
}

// ---------------------------------------------------------------------------
// compound partial mean: 16 blocks x 256 rows -> part[block][10]
// ---------------------------------------------------------------------------
__global__ __launch_bounds__(256) void compound_part_kernel(
    const float* __restrict__ xs, float* __restrict__ part) {
  __shared__ float sred[256 * 10];
  int tid = threadIdx.x;
  int r = blockIdx.x * 256 + tid;
#pragma unroll
  for (int d = 0; d < 10; ++d)
    sred[tid * 10 + d] = xs[r * DPAD + d] * (1.0f / (float)NATOMS);
  __syncthreads();
  for (int s = 128; s > 0; s >>= 1) {
    if (tid < s)
#pragma unroll
      for (int d = 0; d < 10; ++d)
        sred[tid * 10 + d] += sred[(tid + s) * 10 + d];
    __syncthreads();
  }
  if (tid < 10) part[blockIdx.x * 10 + tid] = sred[tid];
}

// deterministic column-sum of [nb,10] partials
__global__ void reduce_cols_kernel(const float* __restrict__ part, int nb,
                                   float* __restrict__ outv) {
  int d = threadIdx.x;
  if (d < 10) {
    float acc = 0.0f;
    for (int b = 0; b < nb; ++b) acc += part[b * 10 + d];
    outv[d] = acc;
  }
}

// ---------------------------------------------------------------------------
// 23x23 "image" conv over [L,10] (pad 11 both dims), relu.
// block = 16 rows x 16 cols; input halo staged to LDS (async on gfx1250)
// ---------------------------------------------------------------------------
__global__ __launch_bounds__(256) void conv_kernel(
    const float* __restrict__ in, const float* __restrict__ wk,
    const float* __restrict__ bptr, float* __restrict__ out) {
  __shared__ __align__(16) float shIn[38 * 16];
  __shared__ float shW[529];
  const int tid = threadIdx.x;
  const int l0  = blockIdx.x * 16;

#if defined(HAVE_ASYNC_LDS)
  // 38 rows x 64B, one 16B async chunk per lane (152 chunks).
  if (tid < 152) {
    int r = tid >> 2, part = tid & 3;
    int gr = l0 - 11 + r;
    if (gr >= 0 && gr < LWORDS) {
      const float* gsrc = in + gr * 16 + part * 4;
      __builtin_amdgcn_global_load_async_to_lds_b128(
          (gptr_v4i)(float*)gsrc,
          (lptr_v4i)(shIn + tid * 4),
          /*offset=*/0, /*cpol=*/0);
    } else {
      // halo outside the image: zero-fill (disjoint addresses, no race)
      shIn[tid * 4 + 0] = 0.0f;
      shIn[tid * 4 + 1] = 0.0f;
      shIn[tid * 4 + 2] = 0.0f;
      shIn[tid * 4 + 3] = 0.0f;
    }
  }
#else
  for (int i = tid; i < 38 * 16; i += 256) {
    int r = i >> 4, cc = i & 15;
    int gr = l0 - 11 + r;
    shIn[i] = (gr >= 0 && gr < LWORDS) ? in[gr * 16 + cc] : 0.0f;
  }
#endif
  for (int i = tid; i < 529; i += 256) shW[i] = wk[i];
  wait_asynccnt0();
  __syncthreads();

  int tl = tid >> 4, d = tid & 15;
  int dd = (d < 10) ? d : 0;                  // keep LDS index in range
  float acc = bptr[0];
  for (int ky = 0; ky < 23; ++ky) {
    const float* srow = &shIn[(tl + ky) * 16];
    const float* wrow = &shW[ky * 23 + 11 - dd];
#pragma unroll
    for (int cc = 0; cc < 10; ++cc) acc += srow[cc] * wrow[cc];
  }
  out[(l0 + tl) * 16 + d] = (d < 10) ? fmaxf(acc, 0.0f) : 0.0f;
}

// h = relu(compound @ Wat^T + bat)
__global__ void h_kernel(const float* __restrict__ compound,
                         const float* __restrict__ Wat,
                         const float* __restrict__ bat,
                         float* __restrict__ h) {
  int d = threadIdx.x;
  if (d < 10) {
    float acc = bat[d];
    for (int e = 0; e < 10; ++e) acc += compound[e] * Wat[d * 10 + e];
    h[d] = fmaxf(acc, 0.0f);
  }
}

// ---------------------------------------------------------------------------
// fused attention: hs_p = relu(xs_p@Wat^T+b); w = tanh(h . hs_p);
// block-partial of mean(w * hs_p)
// ---------------------------------------------------------------------------
__global__ __launch_bounds__(256) void attn_kernel(
    const float* __restrict__ img, const float* __restrict__ Wat,
    const float* __restrict__ bat, const float* __restrict__ h,
    float* __restrict__ part) {
  __shared__ float sred[256 * 10];
  int tid = threadIdx.x;
  int l = blockIdx.x * 256 + tid;
  const float* row = img + l * 16;
  float hp[10];
  float wdot = 0.0f;
#pragma unroll
  for (int d = 0; d < 10; ++d) {
    float acc = bat[d];
#pragma unroll
    for (int e = 0; e < 10; ++e) acc += row[e] * Wat[d * 10 + e];
    hp[d] = fmaxf(acc, 0.0f);
    wdot += h[d] * hp[d];
  }
  float wgt = tanhf(wdot) * (1.0f / (float)LWORDS);
#pragma unroll
  for (int d = 0; d < 10; ++d) sred[tid * 10 + d] = wgt * hp[d];
  __syncthreads();
  for (int s = 128; s > 0; s >>= 1) {
    if (tid < s)
#pragma unroll
      for (int d = 0; d < 10; ++d)
        sred[tid * 10 + d] += sred[(tid + s) * 10 + d];
    __syncthreads();
  }
  if (tid < 10) part[blockIdx.x * 10 + tid] = sred[tid];
}

// fusion MLP (tiny) -> d_out[2]
__global__ void final_kernel(const float* __restrict__ compound,
                             const float* __restrict__ protein,
                             const float* __restrict__ Wout,
                             const float* __restrict__ bout,
                             const float* __restrict__ Wint,
                             const float* __restrict__ bint,
                             float* __restrict__ out) {
  if (threadIdx.x != 0 || blockIdx.x != 0) return;
  float cat[20], nxt[20];
  for (int d = 0; d < 10; ++d) { cat[d] = compound[d]; cat[10 + d] = protein[d]; }
  for (int j = 0; j < 3; ++j) {
    for (int o = 0; o < 20; ++o) {
      float acc = bout[j * 20 + o];
      for (int e = 0; e < 20; ++e) acc += cat[e] * Wout[j * 400 + o * 20 + e];
      nxt[o] = fmaxf(acc, 0.0f);
    }
    for (int o = 0; o < 20; ++o) cat[o] = nxt[o];
  }
  for (int o = 0; o < 2; ++o) {
    float acc = bint[o];
    for (int e = 0; e < 20; ++e) acc += cat[e] * Wint[o * 20 + e];
    out[o] = acc;
  }
}

// ---------------------------------------------------------------------------
extern "C" void kernel_launch(void* const* d_in, const int* in_sizes, int n_in,
                              void* d_out, int out_size, void* d_ws, size_t ws_size,
                              hipStream_t stream) {
  (void)in_sizes; (void)n_in; (void)out_size; (void)ws_size;

  const int*   fp     = (const int*)d_in[0];
  const float* adj    = (const float*)d_in[1];
  const int*   words  = (const int*)d_in[2];
  const float* emb_fp = (const float*)d_in[3];
  const float* emb_w  = (const float*)d_in[4];
  const float* Wg     = (const float*)d_in[5];
  const float* bg     = (const float*)d_in[6];
  const float* Wc     = (const float*)d_in[7];
  const float* bc     = (const float*)d_in[8];
  const float* Wat    = (const float*)d_in[9];
  const float* bat    = (const float*)d_in[10];
  const float* Wout   = (const float*)d_in[11];
  const float* bout   = (const float*)d_in[12];
  const float* Wint   = (const float*)d_in[13];
  const float* bint   = (const float*)d_in[14];
  float* out = (float*)d_out;
  float* ws  = (float*)d_ws;

  // workspace layout (floats)
  float* xs       = ws;                                    // 4096*16
  float* hs       = ws + NATOMS * DPAD;                    // 4096*16
  float* img0     = ws + 2 * NATOMS * DPAD;                // 65536*16
  float* img1     = img0 + LWORDS * DPAD;                  // 65536*16
  float* compound = img1 + LWORDS * DPAD;                  // 16
  float* hvec     = compound + 16;                         // 16
  float* protein  = compound + 32;                         // 16
  float* cpart    = compound + 48;                         // 16*10
  float* ppart    = cpart + 160;                           // 256*10

  // ---- GNN ----
  embed_fp_kernel<<<NATOMS * DPAD / 256, 256, 0, stream>>>(fp, emb_fp, xs);
  for (int layer = 0; layer < 3; ++layer) {
    gnn_hs_kernel<<<NATOMS * DPAD / 256, 256, 0, stream>>>(
        xs, Wg + layer * 100, bg + layer * 10, hs);
    gnn_gemm_kernel<<<NATOMS / 16, 256, 0, stream>>>(adj, hs, xs);
  }
  compound_part_kernel<<<NATOMS / 256, 256, 0, stream>>>(xs, cpart);
  reduce_cols_kernel<<<1, 32, 0, stream>>>(cpart, NATOMS / 256, compound);

  // ---- CNN ----
  embed_word_kernel<<<LWORDS * DPAD / 256, 256, 0, stream>>>(words, emb_w, img0);
  conv_kernel<<<LWORDS / 16, 256, 0, stream>>>(img0, Wc + 0 * 529, bc + 0, img1);
  conv_kernel<<<LWORDS / 16, 256, 0, stream>>>(img1, Wc + 1 * 529, bc + 1, img0);
  conv_kernel<<<LWORDS / 16, 256, 0, stream>>>(img0, Wc + 2 * 529, bc + 2, img1);

  // ---- attention ----
  h_kernel<<<1, 32, 0, stream>>>(compound, Wat, bat, hvec);
  attn_kernel<<<LWORDS / 256, 256, 0, stream>>>(img1, Wat, bat, hvec, ppart);
  reduce_cols_kernel<<<1, 32, 0, stream>>>(ppart, LWORDS / 256, protein);

  // ---- fusion MLP ----
  final_kernel<<<1, 1, 0, stream>>>(compound, protein, Wout, bout, Wint, bint, out);
}